// SelfOccluMask_34368328303287
// MI455X (gfx1250) — compile-verified
//
#include <hip/hip_runtime.h>

#ifndef __has_builtin
#define __has_builtin(x) 0
#endif

namespace {
constexpr int kMaxDisp = 21;
constexpr int kH  = 384;
constexpr int kW  = 1280;
constexpr int kPX = 5;    // pixels per thread along W
constexpr int kNT = 256;  // threads per block (8 waves)
constexpr int kTH = 8;    // output rows per block (band): 48*8 = 384 blocks
constexpr int kWin = kPX + kMaxDisp + 1;  // 27-wide register window of s-values
static_assert(kPX * kNT == kW, "one block covers a full image row");
static_assert(kH % kTH == 0, "bands tile H");
}

// Async copy of 16 bytes global -> LDS (CDNA5 GLOBAL_LOAD_ASYNC_TO_LDS_B128,
// tracked by ASYNCcnt). Inline asm so it works on both toolchains.
__device__ __forceinline__ void async_copy_b128(const float* gsrc, float* lds_dst) {
    // Flat LDS address low 32 bits == LDS byte offset (ISA 10.2 aperture rules).
    unsigned ldsaddr = (unsigned)(unsigned long long)(void*)lds_dst;
    asm volatile("global_load_async_to_lds_b128 %0, %1, off"
                 :: "v"(ldsaddr), "v"(gsrc)
                 : "memory");
}

__device__ __forceinline__ void wait_async0() {
#if __has_builtin(__builtin_amdgcn_s_wait_asynccnt)
    __builtin_amdgcn_s_wait_asynccnt(0);
#else
    asm volatile("s_wait_asynccnt 0" ::: "memory");
#endif
}

// Stage one full 1280-float row into an LDS ring slot: 256 lanes * 16B + 64 lanes * 16B.
__device__ __forceinline__ void issue_row(const float* img, int grow, float* dstrow, int t) {
    const float* src = img + (size_t)grow * kW;
    async_copy_b128(src + 4 * t, dstrow + 4 * t);
    if (t < (kW / 4 - kNT)) {  // remaining 64 chunks of 16B
        async_copy_b128(src + 4 * (kNT + t), dstrow + 4 * (kNT + t));
    }
}

__global__ __launch_bounds__(kNT)
void selfocclu_kernel(const float* __restrict__ disp,
                      const float* __restrict__ bsline,
                      float* __restrict__ out)
{
    __shared__ float ring[4][kW];  // rotating raw rows (async-filled)
    __shared__ float srow[kW];     // 3-row vertical sums for current h

    const int t  = threadIdx.x;
    const int n  = blockIdx.z;
    const int h0 = blockIdx.y * kTH;
    const int p0 = t * kPX;

    const float bs   = bsline[n];
    const float* img = disp + (size_t)n * kH * kW;
    float* outn      = out  + (size_t)n * kH * kW;

    // bs is uniform per block -> force scalar branches
    const int zeroPath = __builtin_amdgcn_readfirstlane(bs == 0.0f ? 1 : 0);
    const int lookR    = __builtin_amdgcn_readfirstlane(bs <  0.0f ? 1 : 0);  // maskl

    if (zeroPath) {
        for (int h = h0; h < h0 + kTH; ++h) {
            float* o = outn + (size_t)h * kW + p0;
#pragma unroll
            for (int k = 0; k < kPX; ++k) o[k] = 0.0f;
        }
        return;
    }

    // Prologue: stage logical rows h0-1, h0, h0+1 (row-clamped); slot(r) = (r+1)&3
#pragma unroll
    for (int r = -1; r <= 1; ++r) {
        int lr = h0 + r;
        int gr = lr < 0 ? 0 : (lr > kH - 1 ? kH - 1 : lr);
        issue_row(img, gr, ring[(lr + 1) & 3], t);
    }
    wait_async0();
    __syncthreads();

    for (int h = h0; h < h0 + kTH; ++h) {
        // Prefetch logical row h+2 into slot (h+3)&3 (distinct from h-1,h,h+1 slots);
        // overlaps with this iteration's compute, waited at loop end.
        int rn = h + 2;
        if (rn <= h0 + kTH) {
            int gr = rn > kH - 1 ? kH - 1 : rn;
            issue_row(img, gr, ring[(rn + 1) & 3], t);
        }

        // Build vertical 3-row sums for row h from ring slots h-1,h,h+1.
        {
            const float* r0 = ring[(h + 0) & 3];
            const float* r1 = ring[(h + 1) & 3];
            const float* r2 = ring[(h + 2) & 3];
#pragma unroll
            for (int k = 0; k < kPX; ++k) {
                int c = p0 + k;
                srow[c] = r0[c] + r1[c] + r2[c];
            }
        }
        __syncthreads();

        // 27-wide register window of s-values (stride-5 lane access: conflict-free).
        float win[kWin];
        if (lookR) {
#pragma unroll
            for (int j = 0; j < kWin; ++j) {
                int c = p0 + j;
                win[j] = srow[c < kW ? c : kW - 1];
            }
        } else {
#pragma unroll
            for (int j = 0; j < kWin; ++j) {
                int c = p0 + j - (kMaxDisp + 1);
                win[j] = srow[c > 0 ? c : 0];
            }
        }

        float* o = outn + (size_t)h * kW + p0;
#pragma unroll
        for (int k = 0; k < kPX; ++k) {
            float a0, a1;
            if (lookR) { a0 = win[k];      a1 = win[k + 1];  }
            else       { a0 = win[k + 22]; a1 = win[k + 21]; }
            const float A = (a0 + a1) * (1.0f / 6.0f);
            float m = 3.0e38f;
#pragma unroll
            for (int i = 0; i < kMaxDisp; ++i) {
                float s = lookR ? win[k + 2 + i] : win[k + 20 - i];
                float v = fabsf(fmaf(s, -(1.0f / 3.0f), A + (float)(i + 1)));
                m = fminf(m, v);  // |x| is a free src modifier; pairs fold to v_min3
            }
            o[k] = fminf(m, 1.0f);  // m >= 0 already
        }

        wait_async0();     // prefetched row h+2 landed (own wave)
        __syncthreads();   // ...for all waves; also srow WAR for next iteration
    }
}

extern "C" void kernel_launch(void* const* d_in, const int* in_sizes, int n_in,
                              void* d_out, int out_size, void* d_ws, size_t ws_size,
                              hipStream_t stream) {
    (void)n_in; (void)out_size; (void)d_ws; (void)ws_size;
    const float* disp = (const float*)d_in[0];
    const float* bs   = (const float*)d_in[1];
    float* out        = (float*)d_out;
    const int N = in_sizes[1];  // 8
    dim3 grid(1, kH / kTH, N);
    dim3 block(kNT);
    selfocclu_kernel<<<grid, block, 0, stream>>>(disp, bs, out);
}